// Model_21809843929738
// MI455X (gfx1250) — compile-verified
//
#include <hip/hip_runtime.h>

typedef __attribute__((ext_vector_type(2))) float v2f;
typedef __attribute__((ext_vector_type(8))) float v8f;

#define CHUNK       64                    // sequence elements per chunk (E)
#define TILE_CHUNKS 16                    // chunks per WMMA tile (M/N dim)
#define TILE_ELEMS  (CHUNK * TILE_CHUNKS) // 1024 elements = 4 KB per tile
#define KSTEPS      (CHUNK / 4)           // 16 wmma k-steps (K=4 each)
#define NPOW        18                    // covers exponents < 2^18 chunks

// workspace layout (in floats)
#define WS_H    0                         // H[64][4]          : 256 floats
#define WS_POW  256                       // (M^T)^{64*2^p}    : 18*16 floats
#define WS_R    (WS_POW + NPOW * 16)      // per-tile-step mtx : 16 floats
#define WS_PART 1024                      // per-block partials

__device__ __forceinline__ void mat4_mul(const float* A, const float* B, float* C) {
#pragma unroll
  for (int r = 0; r < 4; ++r)
#pragma unroll
    for (int c = 0; c < 4; ++c) {
      float s = 0.0f;
#pragma unroll
      for (int k = 0; k < 4; ++k) s += A[r * 4 + k] * B[k * 4 + c];
      C[r * 4 + c] = s;
    }
}

// Single-thread precompute of H, power table, and R = (M^T)^(TILE_ELEMS*nWaves).
__global__ void lrec_setup(const float* __restrict__ W, float* __restrict__ ws, int nWaves) {
  if (threadIdx.x != 0 || blockIdx.x != 0) return;
  // v' = M v + x b ;  M[n][k] = (k==0 ? 0 : W[k][n]) ; b[n] = W[0][n]
  float M[16], b[4];
  for (int n = 0; n < 4; ++n) {
    b[n] = W[n];
    for (int k = 0; k < 4; ++k) M[n * 4 + k] = (k == 0) ? 0.0f : W[k * 4 + n];
  }
  // H[j][:] = M^(63-j) b
  float w[4] = {b[0], b[1], b[2], b[3]};
  for (int j = CHUNK - 1; j >= 0; --j) {
    for (int m = 0; m < 4; ++m) ws[WS_H + j * 4 + m] = w[m];
    float t[4];
    for (int n = 0; n < 4; ++n) {
      float s = 0.0f;
      for (int k = 0; k < 4; ++k) s += M[n * 4 + k] * w[k];
      t[n] = s;
    }
    for (int n = 0; n < 4; ++n) w[n] = t[n];
  }
  // S = (M^T)^64 via 6 squarings
  float S[16], T[16];
  for (int r = 0; r < 4; ++r)
    for (int c = 0; c < 4; ++c) S[r * 4 + c] = M[c * 4 + r];
  for (int it = 0; it < 6; ++it) { mat4_mul(S, S, T); for (int i = 0; i < 16; ++i) S[i] = T[i]; }
  float P[NPOW][16];
  for (int i = 0; i < 16; ++i) P[0][i] = S[i];
  for (int p = 1; p < NPOW; ++p) mat4_mul(P[p - 1], P[p - 1], P[p]);
  for (int p = 0; p < NPOW; ++p)
    for (int i = 0; i < 16; ++i) ws[WS_POW + p * 16 + i] = P[p][i];
  // R = (M^T)^(64 * TILE_CHUNKS * nWaves); exponent in (M^T)^64 units:
  float R[16] = {1,0,0,0, 0,1,0,0, 0,0,1,0, 0,0,0,1};
  unsigned eu = (unsigned)TILE_CHUNKS * (unsigned)nWaves;
  for (int p = 0; p < NPOW; ++p) {
    if ((eu >> p) & 1u) { float C[16]; mat4_mul(P[p], R, C); for (int i = 0; i < 16; ++i) R[i] = C[i]; }
  }
  for (int i = 0; i < 16; ++i) ws[WS_R + i] = R[i];
}

// Streaming kernel: per wave, tiles in decreasing order; 16 x v_wmma_f32_16x16x4_f32
// per 4 KB tile; per-lane g-vector advanced by one 4x4 matvec per tile.
__global__ void __launch_bounds__(256) lrec_phase1(const float* __restrict__ seq,
                                                   const float* __restrict__ ws,
                                                   float* __restrict__ partials,
                                                   int N, int nWaves) {
  const int tid  = threadIdx.x;
  const int lane = tid & 31;
  const int n    = lane & 15;   // tile row/col index
  const int hi   = lane >> 4;   // K half (0 -> K=0,1 ; 1 -> K=2,3)
  const int waveId = (int)((blockIdx.x * blockDim.x + tid) >> 5);

  const int nChunks = N / CHUNK;
  const int nTiles  = N / TILE_ELEMS;

  // Preload constant A operands (H^T slices) for all 16 k-steps.
  // A(16x4): row M = lane%16 (rows >=4 are zero), lane half selects K pair.
  v2f aReg[KSTEPS];
#pragma unroll
  for (int s = 0; s < KSTEPS; ++s) {
    const int j = 4 * s + 2 * hi;
    v2f a = {0.0f, 0.0f};
    if (n < 4) { a[0] = ws[WS_H + j * 4 + n]; a[1] = ws[WS_H + (j + 1) * 4 + n]; }
    aReg[s] = a;
  }
  float Rm[16];
#pragma unroll
  for (int i = 0; i < 16; ++i) Rm[i] = ws[WS_R + i];

  float localSum = 0.0f;
  if (waveId < nTiles) {
    const int Tlast = waveId + ((nTiles - 1 - waveId) / nWaves) * nWaves;
    // Seed g = (M^T)^(nChunks-1-chunkIdx) e1 via binary power table.
    float g0 = 0.0f, g1 = 1.0f, g2 = 0.0f, g3 = 0.0f;
    const unsigned e = (unsigned)(nChunks - 1 - (Tlast * TILE_CHUNKS + n));
#pragma unroll
    for (int p = 0; p < NPOW; ++p) {
      if ((e >> p) & 1u) {
        const float* Pm = ws + WS_POW + p * 16;
        const float h0 = Pm[ 0]*g0 + Pm[ 1]*g1 + Pm[ 2]*g2 + Pm[ 3]*g3;
        const float h1 = Pm[ 4]*g0 + Pm[ 5]*g1 + Pm[ 6]*g2 + Pm[ 7]*g3;
        const float h2 = Pm[ 8]*g0 + Pm[ 9]*g1 + Pm[10]*g2 + Pm[11]*g3;
        const float h3 = Pm[12]*g0 + Pm[13]*g1 + Pm[14]*g2 + Pm[15]*g3;
        g0 = h0; g1 = h1; g2 = h2; g3 = h3;
      }
    }
    for (int T = Tlast; T >= 0; T -= nWaves) {
      const float* base = seq + (size_t)T * TILE_ELEMS + (size_t)n * CHUNK + 2 * hi;
      if (T >= nWaves)  // speculative prefetch of next tile (global_prefetch_b8)
        __builtin_prefetch(base - (size_t)nWaves * TILE_ELEMS, 0, 1);
      v8f acc = {};
#pragma unroll
      for (int s = 0; s < KSTEPS; ++s) {
        const v2f bv = *(const v2f*)(base + 4 * s);
        // D(16x16) += A(16x4, H^T slice) x B(4x16, data slice)
        acc = __builtin_amdgcn_wmma_f32_16x16x4_f32(false, aReg[s], false, bv,
                                                    (short)0, acc, false, false);
      }
      // Lane n<16: acc[0..3] = s_chunk (D rows 0..3); upper-half lanes hold zero rows.
      localSum += g0 * acc[0] + g1 * acc[1] + g2 * acc[2] + g3 * acc[3];
      // Advance g to the next (lower) tile: exponent += TILE_CHUNKS*nWaves chunk-steps.
      const float h0 = Rm[ 0]*g0 + Rm[ 1]*g1 + Rm[ 2]*g2 + Rm[ 3]*g3;
      const float h1 = Rm[ 4]*g0 + Rm[ 5]*g1 + Rm[ 6]*g2 + Rm[ 7]*g3;
      const float h2 = Rm[ 8]*g0 + Rm[ 9]*g1 + Rm[10]*g2 + Rm[11]*g3;
      const float h3 = Rm[12]*g0 + Rm[13]*g1 + Rm[14]*g2 + Rm[15]*g3;
      g0 = h0; g1 = h1; g2 = h2; g3 = h3;
    }
  }

  // Deterministic block reduction.
  __shared__ float red[256];
  red[tid] = localSum;
  __syncthreads();
  if (tid == 0) {
    float s = 0.0f;
    for (int i = 0; i < (int)blockDim.x; ++i) s += red[i];
    partials[blockIdx.x] = s;
  }
}

__global__ void lrec_finalize(const float* __restrict__ partials, int nBlocks,
                              float* __restrict__ out) {
  if (threadIdx.x == 0 && blockIdx.x == 0) {
    float s = 0.0f;
    for (int i = 0; i < nBlocks; ++i) s += partials[i];
    out[0] = s;
  }
}

extern "C" void kernel_launch(void* const* d_in, const int* in_sizes, int n_in,
                              void* d_out, int out_size, void* d_ws, size_t ws_size,
                              hipStream_t stream) {
  (void)n_in; (void)out_size; (void)ws_size;
  const float* seq = (const float*)d_in[0];
  const float* W   = (const float*)d_in[1];
  float* out = (float*)d_out;
  float* ws  = (float*)d_ws;
  const int N = in_sizes[0];          // 1<<24, multiple of TILE_ELEMS
  const int blocks = 256, threads = 256;
  const int nWaves = blocks * threads / 32;  // 2048 waves, 8 tiles each

  lrec_setup<<<1, 1, 0, stream>>>(W, ws, nWaves);
  lrec_phase1<<<blocks, threads, 0, stream>>>(seq, ws, ws + WS_PART, N, nWaves);
  lrec_finalize<<<1, 1, 0, stream>>>(ws + WS_PART, blocks, out);
}